// MultiBoxLoss_22153441313260
// MI455X (gfx1250) — compile-verified
//
#include <hip/hip_runtime.h>
#include <stdint.h>

#ifndef __has_builtin
#define __has_builtin(x) 0
#endif

#if defined(__HIP_DEVICE_COMPILE__) && defined(__gfx1250__) && \
    __has_builtin(__builtin_amdgcn_global_load_async_to_lds_b128)
#define USE_ASYNC_LDS 1
#else
#define USE_ASYNC_LDS 0
#endif

#define BATCH   128
#define NBOX    8732
#define CLS     21
#define CHUNK   256
#define NCHUNK  35            /* ceil(8732/256) */
#define N4      2183          /* 8732/4 exact   */
#define BT      512

typedef int v4i __attribute__((vector_size(16)));

// ---------- CDNA5 async global -> LDS helpers (fallback: plain b128 copy) ----------
__device__ __forceinline__ void g2l_b128(const void* g, void* l) {
#if USE_ASYNC_LDS
  // Builtin signature (from hipcc diagnostic): (v4i AS1*, v4i AS3*, imm offset, imm cpol).
  // Global generic<->AS1 addresses are numerically identical, so the integer hop is safe;
  // LDS needs a true addrspacecast.
  __builtin_amdgcn_global_load_async_to_lds_b128(
      (__attribute__((address_space(1))) v4i*)(uintptr_t)g,
      (__attribute__((address_space(3))) v4i*)l, 0, 0);
#else
  *(float4*)l = *(const float4*)g;
#endif
}

__device__ __forceinline__ void g2l_wait() {
#if USE_ASYNC_LDS
#if __has_builtin(__builtin_amdgcn_s_wait_asynccnt)
  __builtin_amdgcn_s_wait_asynccnt(0);
#else
  asm volatile("s_wait_asynccnt 0" ::: "memory");
#endif
#endif
}

// Monotone total-order encoding of float into uint32 (handles the tiny-negative
// CE case from rounding; positives encode as enc(0.0f) = 0x80000000, matching
// the reference which ranks positive boxes at mining value 0).
__device__ __forceinline__ unsigned fenc(float f) {
  unsigned u = __float_as_uint(f);
  return (u & 0x80000000u) ? ~u : (u | 0x80000000u);
}

__device__ __forceinline__ float sl1(float d) {
  float a = fabsf(d);
  return (a < 1.0f) ? 0.5f * d * d : a - 0.5f;
}

// ================= Kernel A: per-box CE + smooth-L1 + per-slab partials =================
__global__ __launch_bounds__(CHUNK) void mb_box_kernel(
    const float* __restrict__ conf, const int* __restrict__ tgt,
    const float* __restrict__ locp, const float* __restrict__ loct,
    float* __restrict__ ceAll, unsigned* __restrict__ ceBits,
    float* __restrict__ locPart, int* __restrict__ posPart) {
  __shared__ __align__(16) float sConf[6 * 1024];  // 24 KB staging: 256 boxes x 21 cls
  __shared__ float sRedF[CHUNK];
  __shared__ int   sRedI[CHUNK];

  const int blk   = blockIdx.x;
  const int row   = blk / NCHUNK;
  const int chunk = blk - row * NCHUNK;
  const int t     = threadIdx.x;
  const int n0    = chunk * CHUNK;

  const long base   = ((long)row * NBOX + n0) * CLS;           // float index
  const long totalF = (long)BATCH * NBOX * CLS;
  const int  needF  = CHUNK * CLS;                             // 5376 floats

  // Async-stage the slab's logits (contiguous) into LDS, 16B per lane per iter.
#pragma unroll
  for (int it = 0; it < 6; ++it) {
    const int fl = (it * CHUNK + t) * 4;
    if (fl < needF) {
      long g = base + fl;
      if (g > totalF - 4) g = totalF - 4;  // clamp at buffer end (tail of last row)
      g2l_b128(conf + g, &sConf[fl]);
    }
  }
  g2l_wait();
  __syncthreads();

  const int n = n0 + t;
  float l1 = 0.0f;
  int posf = 0;
  if (n < NBOX) {
    const float* x = &sConf[t * CLS];   // stride 21 -> conflict-free (gcd(21,64)=1)
    float m = x[0];
#pragma unroll
    for (int c = 1; c < CLS; ++c) m = fmaxf(m, x[c]);
    float s = 0.0f;
#pragma unroll
    for (int c = 0; c < CLS; ++c) s += __expf(x[c] - m);
    const long gi = (long)row * NBOX + n;
    const int  y  = tgt[gi];
    const float ce = __logf(s) + m - x[y];   // exact CE (stabilization choice is moot)
    posf = (y > 0) ? 1 : 0;
    ceAll[gi]  = ce;
    ceBits[gi] = fenc(posf ? 0.0f : ce);
    if (posf) {
      const float4 p = ((const float4*)locp)[gi];
      const float4 q = ((const float4*)loct)[gi];
      l1 = sl1(p.x - q.x) + sl1(p.y - q.y) + sl1(p.z - q.z) + sl1(p.w - q.w);
    }
  }

  // deterministic fixed-tree block reduction
  sRedF[t] = l1;
  sRedI[t] = posf;
  __syncthreads();
  for (int s2 = CHUNK / 2; s2 > 0; s2 >>= 1) {
    if (t < s2) { sRedF[t] += sRedF[t + s2]; sRedI[t] += sRedI[t + s2]; }
    __syncthreads();
  }
  if (t == 0) { locPart[blk] = sRedF[0]; posPart[blk] = sRedI[0]; }
}

// ====== Kernel B: per-row radix-select (hard-neg mining) + masked CE accumulation ======
__global__ __launch_bounds__(BT) void mb_row_kernel(
    const unsigned* __restrict__ ceBits, const float* __restrict__ ceAll,
    const int* __restrict__ tgt,
    const float* __restrict__ locPart, const int* __restrict__ posPart,
    float* __restrict__ rowLoc, float* __restrict__ rowConf, int* __restrict__ rowPos) {
  __shared__ __align__(16) unsigned sBits[NBOX];   // 34.9 KB: whole row of keys
  __shared__ int hist[256];
  __shared__ float sRedF[BT];
  __shared__ int   sRedI[BT];
  __shared__ int   waveTot[BT / 32];
  __shared__ unsigned shPrefix;
  __shared__ int shK, sNumPos, shRun, shIterTot;

  const int row = blockIdx.x;
  const int t   = threadIdx.x;
  const long rbase = (long)row * NBOX;

  // Async-stage the row's encoded keys to LDS (2183 x b128, exact).
  const unsigned* src = ceBits + rbase;
  for (int i = t; i < N4; i += BT) g2l_b128(src + (long)i * 4, &sBits[i * 4]);

  if (t == 0) {
    float ls = 0.f; int pc = 0;
    for (int c = 0; c < NCHUNK; ++c) {        // fixed order -> deterministic
      ls += locPart[row * NCHUNK + c];
      pc += posPart[row * NCHUNK + c];
    }
    rowLoc[row] = ls; rowPos[row] = pc; sNumPos = pc; shRun = 0;
  }
  g2l_wait();
  __syncthreads();

  const int K = min(3 * sNumPos, NBOX - 1);   // uniform across block
  unsigned T = 0xFFFFFFFFu;                   // "nothing passes" sentinel for K==0
  int R = 0;                                  // ties to admit at threshold
  if (K > 0) {
    // 4-pass byte-radix select of the K-th largest key.
    unsigned prefix = 0u;
    int kk = K;
    for (int p = 3; p >= 0; --p) {
      if (t < 256) hist[t] = 0;
      __syncthreads();
      const unsigned hmask = (p == 3) ? 0u : (0xFFFFFFFFu << ((p + 1) * 8));
      for (int i = t; i < NBOX; i += BT) {
        const unsigned v = sBits[i];
        if ((v & hmask) == prefix) atomicAdd(&hist[(v >> (p * 8)) & 255], 1);
      }
      __syncthreads();
      if (t == 0) {
        int k2 = kk, bin = 0;
        for (int v = 255; v >= 0; --v) {
          const int c = hist[v];
          if (k2 <= c) { bin = v; break; }
          k2 -= c;
        }
        shK = k2;
        shPrefix = prefix | ((unsigned)bin << (p * 8));
      }
      __syncthreads();
      prefix = shPrefix;
      kk = shK;
      __syncthreads();
    }
    T = prefix;
    // G = count strictly greater than threshold
    int g = 0;
    for (int i = t; i < NBOX; i += BT) g += (sBits[i] > T) ? 1 : 0;
    sRedI[t] = g;
    __syncthreads();
    for (int s2 = BT / 2; s2 > 0; s2 >>= 1) {
      if (t < s2) sRedI[t] += sRedI[t + s2];
      __syncthreads();
    }
    R = K - sRedI[0];
  }
  __syncthreads();

  // Ordered sweep: admit exactly R threshold-ties in ascending index order
  // (== JAX stable argsort semantics), accumulating masked CE.
  float acc = 0.0f;
  for (int basei = 0; basei < NBOX; basei += BT) {
    const int i = basei + t;
    const int active = (i < NBOX) ? 1 : 0;
    const unsigned v = active ? sBits[i] : 0u;
    const int isTie = (active && v == T) ? 1 : 0;
    const unsigned long long bal = __ballot(isTie);   // wave32: low 32 bits
    const int lane = t & 31, wv = t >> 5;
    if (lane == 0) waveTot[wv] = __popcll(bal);
    __syncthreads();
    if (t == 0) {                      // exclusive scan over the 16 wave totals
      int s = 0;
      for (int w = 0; w < BT / 32; ++w) { int c = waveTot[w]; waveTot[w] = s; s += c; }
      shIterTot = s;
    }
    __syncthreads();
    const int ord = shRun + waveTot[wv] + __popcll(bal & ((1ull << lane) - 1ull));
    const int neg = active && ((v > T) || (isTie && ord < R));
    if (active) {
      const int y = tgt[rbase + i];
      if ((y > 0) || neg) acc += ceAll[rbase + i];
    }
    __syncthreads();
    if (t == 0) shRun += shIterTot;
  }

  sRedF[t] = acc;
  __syncthreads();
  for (int s2 = BT / 2; s2 > 0; s2 >>= 1) {
    if (t < s2) sRedF[t] += sRedF[t + s2];
    __syncthreads();
  }
  if (t == 0) rowConf[row] = sRedF[0];
}

// ================= Kernel C: deterministic final scalar =================
__global__ void mb_final_kernel(const float* __restrict__ rowLoc,
                                const float* __restrict__ rowConf,
                                const int* __restrict__ rowPos,
                                float* __restrict__ out) {
  if (threadIdx.x == 0 && blockIdx.x == 0) {
    float l = 0.f, c = 0.f; int m = 0;
    for (int r = 0; r < BATCH; ++r) { l += rowLoc[r]; c += rowConf[r]; m += rowPos[r]; }
    out[0] = (m > 0) ? (l + c) / (float)m : 0.0f;
  }
}

extern "C" void kernel_launch(void* const* d_in, const int* in_sizes, int n_in,
                              void* d_out, int out_size, void* d_ws, size_t ws_size,
                              hipStream_t stream) {
  (void)in_sizes; (void)n_in; (void)out_size; (void)ws_size;
  const float* locp = (const float*)d_in[0];
  const float* loct = (const float*)d_in[1];
  const float* conf = (const float*)d_in[2];
  const int*   tgt  = (const int*)d_in[3];
  float* out = (float*)d_out;

  char* ws = (char*)d_ws;
  size_t off = 0;
  auto take = [&](size_t bytes) {
    void* p = ws + off;
    off += (bytes + 255) & ~(size_t)255;
    return p;
  };
  float*    ceAll   = (float*)take((size_t)BATCH * NBOX * sizeof(float));
  unsigned* ceBits  = (unsigned*)take((size_t)BATCH * NBOX * sizeof(unsigned));
  float*    locPart = (float*)take((size_t)BATCH * NCHUNK * sizeof(float));
  int*      posPart = (int*)take((size_t)BATCH * NCHUNK * sizeof(int));
  float*    rowLoc  = (float*)take(BATCH * sizeof(float));
  float*    rowConf = (float*)take(BATCH * sizeof(float));
  int*      rowPos  = (int*)take(BATCH * sizeof(int));

  hipLaunchKernelGGL(mb_box_kernel, dim3(BATCH * NCHUNK), dim3(CHUNK), 0, stream,
                     conf, tgt, locp, loct, ceAll, ceBits, locPart, posPart);
  hipLaunchKernelGGL(mb_row_kernel, dim3(BATCH), dim3(BT), 0, stream,
                     ceBits, ceAll, tgt, locPart, posPart, rowLoc, rowConf, rowPos);
  hipLaunchKernelGGL(mb_final_kernel, dim3(1), dim3(32), 0, stream,
                     rowLoc, rowConf, rowPos, out);
}